// sccnn_conv_8976481648962
// MI455X (gfx1250) — compile-verified
//
#include <hip/hip_runtime.h>
#include <stdint.h>

// ---------------------------------------------------------------------------
// SCCNN forward for MI455X (gfx1250, wave32).
// All heavy ops are (n x n) @ (n x 32) fp32 GEMMs -> V_WMMA_F32_16X16X4_F32.
// Memory-bound (16 FLOP/byte fp32 => HBM-limited at 23.3 TB/s): L matrices
// stream once from HBM, repeated Chebyshev passes hit the 192MB L2.
// Steady-state tiles are staged with GLOBAL_LOAD_ASYNC_TO_LDS_B128 (ASYNCcnt)
// into double-buffered LDS, overlapping the next tile's loads with WMMAs.
// ---------------------------------------------------------------------------

typedef float v2f __attribute__((ext_vector_type(2)));
typedef float v8f __attribute__((ext_vector_type(8)));

#define F_DIM    32
#define MBLK     64
#define KBLK     32
#define LPAD     36   // 144B row stride: 16B-aligned for async B128, 16-bank-clean
#define NTHREADS 256

// MODE 0: Y = (aux? aux[m]:1) * (A@X)            (aux = rowscale or null)
// MODE 1: Y = 2*(A@X) - aux[m,f]                 (Chebyshev step, aux = T_{j-2})
// TRANS : A is stored (K x M) row-major; compute with A^T (B1^T, B2^T cases)
template <int MODE, bool TRANS>
__global__ __launch_bounds__(NTHREADS)
void sccnn_gemm32_wmma(const float* __restrict__ A, const float* __restrict__ X,
                       float* __restrict__ Y, const float* __restrict__ aux,
                       int M, int K) {
  __shared__ float As[2][MBLK][LPAD];
  __shared__ float Xs[2][KBLK][LPAD];

  const int tid  = threadIdx.x;
  const int lane = tid & 31;
  const int wid  = tid >> 5;
  const int l15  = lane & 15;
  const int half = lane >> 4;        // 0: lanes 0-15, 1: lanes 16-31
  const int rt   = wid & 3;          // row-tile within block (0..3)
  const int c0   = (wid >> 2) * 16;  // column half (0 or 16)
  const int m0   = blockIdx.x * MBLK;

  v8f acc = {};

  // ---- tile staging: async fast path (full K tile, no transpose),
  //      synchronous fallback (transpose and/or K tail with zero-fill) ----
  auto stage = [&](int buf, int kg0) {
    const bool full_k = (kg0 + KBLK) <= K;
    if (!TRANS && full_k) {
      // A tile: 64x32 = 512 float4 -> 2 async B128 copies per thread
#pragma unroll
      for (int it = 0; it < 2; ++it) {
        int idx = tid + it * NTHREADS;
        int r   = idx >> 3;
        int c4  = (idx & 7) << 2;
        int mg  = m0 + r; if (mg >= M) mg = M - 1;      // clamp: tail rows unused
        const float* gp = A + (size_t)mg * K + (kg0 + c4);
        uint32_t lds = (uint32_t)(uintptr_t)&As[buf][r][c4];
        asm volatile("global_load_async_to_lds_b128 %0, %1, off"
                     :: "v"(lds), "v"(gp) : "memory");
      }
      // X tile: 32x32 = 256 float4 -> 1 async B128 copy per thread
      {
        int kr = tid >> 3;
        int f4 = (tid & 7) << 2;
        const float* gp = X + (size_t)(kg0 + kr) * F_DIM + f4;
        uint32_t lds = (uint32_t)(uintptr_t)&Xs[buf][kr][f4];
        asm volatile("global_load_async_to_lds_b128 %0, %1, off"
                     :: "v"(lds), "v"(gp) : "memory");
      }
      return;
    }
    // ---------- synchronous fallback ----------
    if (!TRANS) {
#pragma unroll
      for (int it = 0; it < 2; ++it) {
        int idx = tid + it * NTHREADS;
        int r   = idx >> 3;
        int c4  = (idx & 7) << 2;
        int mg  = m0 + r; if (mg >= M) mg = M - 1;
#pragma unroll
        for (int j = 0; j < 4; ++j) {
          int kk = kg0 + c4 + j;
          As[buf][r][c4 + j] = (kk < K) ? A[(size_t)mg * K + kk] : 0.0f;
        }
      }
    } else {
      // A^T: global G is (K x M); read contiguous along m, transpose into LDS.
#pragma unroll
      for (int it = 0; it < 2; ++it) {
        int idx = tid + it * NTHREADS;
        int kr  = idx >> 4;
        int m4  = (idx & 15) << 2;
        int kg  = kg0 + kr; if (kg >= K) kg = K - 1;    // X rows past K zeroed
        int mg  = m0 + m4;
        if (mg + 3 < M && kg0 + kr < K) {
          const float4 v = *(const float4*)(A + (size_t)kg * M + mg);
          As[buf][m4 + 0][kr] = v.x; As[buf][m4 + 1][kr] = v.y;
          As[buf][m4 + 2][kr] = v.z; As[buf][m4 + 3][kr] = v.w;
        } else {
#pragma unroll
          for (int j = 0; j < 4; ++j) {
            int mm = mg + j;
            As[buf][m4 + j][kr] = (mm < M) ? A[(size_t)kg * M + mm] : 0.0f;
          }
        }
      }
    }
    {
      int kr = tid >> 3;
      int f4 = (tid & 7) << 2;
      int kg = kg0 + kr;
      float vx = 0.f, vy = 0.f, vz = 0.f, vw = 0.f;
      if (kg < K) {
        const float4 v = *(const float4*)(X + (size_t)kg * F_DIM + f4);
        vx = v.x; vy = v.y; vz = v.z; vw = v.w;
      }
      Xs[buf][kr][f4 + 0] = vx; Xs[buf][kr][f4 + 1] = vy;
      Xs[buf][kr][f4 + 2] = vz; Xs[buf][kr][f4 + 3] = vw;
    }
  };

  const int am = rt * 16 + l15;
  const int nk = (K + KBLK - 1) / KBLK;
  int buf = 0;

  stage(0, 0);
  asm volatile("s_wait_asynccnt 0x0" ::: "memory");
  __syncthreads();

  for (int kt = 0; kt < nk; ++kt) {
    const bool more = (kt + 1) < nk;
    if (more) stage(buf ^ 1, (kt + 1) * KBLK);   // prefetch next tile (async)

    // ---- 8x V_WMMA_F32_16X16X4_F32 over this K tile (full EXEC) ----
    // A frag (16x4 f32): lanes 0-15 K={ks,ks+1}, lanes 16-31 K={ks+2,ks+3}
    // B frag (4x16 f32): mirrored row pairing, N = c0 + (lane & 15)
#pragma unroll
    for (int ks = 0; ks < KBLK; ks += 4) {
      v2f a, b;
      a.x = As[buf][am][ks + 2 * half];
      a.y = As[buf][am][ks + 2 * half + 1];
      b.x = Xs[buf][ks + 2 * half][c0 + l15];
      b.y = Xs[buf][ks + 2 * half + 1][c0 + l15];
      acc = __builtin_amdgcn_wmma_f32_16x16x4_f32(false, a, false, b,
                                                  (short)0, acc, false, false);
    }

    if (more) {
      asm volatile("s_wait_asynccnt 0x0" ::: "memory");
      __syncthreads();
      buf ^= 1;
    }
  }

  // ---- epilogue: C/D layout -> VGPR r: M=r (lanes 0-15) / M=r+8 (16-31) ----
  const int row_base = m0 + rt * 16 + (half ? 8 : 0);
  const int col = c0 + l15;
#pragma unroll
  for (int r = 0; r < 8; ++r) {
    const int mg = row_base + r;
    if (mg < M) {
      float v = acc[r];
      if (MODE == 1) {
        v = 2.0f * v - aux[(size_t)mg * F_DIM + col];   // T_j = 2 L T_{j-1} - T_{j-2}
      } else {
        if (aux) v *= aux[mg];                          // diagonal row scale
      }
      Y[(size_t)mg * F_DIM + col] = v;
    }
  }
}

// Extract diag of a dense (n x n) matrix. mode 0: copy, 1: 1/x, 2: safe 1/x.
__global__ void sccnn_diag_kernel(const float* __restrict__ D,
                                  float* __restrict__ out, int n, int mode) {
  const int i = blockIdx.x * blockDim.x + threadIdx.x;
  if (i < n) {
    const float v = D[(size_t)i * n + i];
    float r;
    if (mode == 0)      r = v;
    else if (mode == 1) r = 1.0f / v;
    else                r = (v != 0.0f) ? (1.0f / v) : 0.0f;
    out[i] = r;
  }
}

// out[n,f] = in[n,f] * s[n]  (pre-scales RHS so GEMM tiles are pure copies)
__global__ void sccnn_rowscale_kernel(const float* __restrict__ in,
                                      const float* __restrict__ s,
                                      float* __restrict__ out, int N) {
  const int i = blockIdx.x * blockDim.x + threadIdx.x;
  if (i < N * F_DIM) out[i] = in[i] * s[i >> 5];
}

// y[n,o] = relu( sum_k sum_i X_k[n,i] * W[i,o,k] ), terms: (KT, N, 32),
// W: (32, 32, KT) contiguous, cached in LDS (<= 45KB << 320KB/WGP).
__global__ __launch_bounds__(NTHREADS)
void sccnn_fused_out_kernel(const float* __restrict__ terms,
                            const float* __restrict__ W,
                            float* __restrict__ y, int N, int KT) {
  __shared__ float Ws[32 * 32 * 11];
  const int wtot = 32 * 32 * KT;
  for (int idx = threadIdx.x; idx < wtot; idx += NTHREADS) Ws[idx] = W[idx];
  __syncthreads();

  const int n = blockIdx.x * (NTHREADS / 32) + (threadIdx.x >> 5);
  const int o = threadIdx.x & 31;
  if (n < N) {
    float acc = 0.0f;
    for (int k = 0; k < KT; ++k) {
      const float* Xk = terms + (size_t)k * N * 32 + (size_t)n * 32;
#pragma unroll
      for (int i = 0; i < 32; ++i)
        acc = fmaf(Xk[i], Ws[i * 32 * KT + o * KT + k], acc);
    }
    y[(size_t)n * 32 + o] = fmaxf(acc, 0.0f);
  }
}

// ---------------------------------------------------------------------------

template <int MODE, bool TRANS>
static inline void run_gemm(const float* A, const float* X, float* Y,
                            const float* aux, int M, int K, hipStream_t s) {
  dim3 grid((M + MBLK - 1) / MBLK), block(NTHREADS);
  hipLaunchKernelGGL((sccnn_gemm32_wmma<MODE, TRANS>), grid, block, 0, s,
                     A, X, Y, aux, M, K);
}

extern "C" void kernel_launch(void* const* d_in, const int* in_sizes, int n_in,
                              void* d_out, int out_size, void* d_ws, size_t ws_size,
                              hipStream_t stream) {
  (void)in_sizes; (void)n_in; (void)out_size; (void)ws_size;
  constexpr int N0 = 1500, N1 = 6000, N2 = 4000;

  const float* x0  = (const float*)d_in[0];
  const float* x1  = (const float*)d_in[1];
  const float* x2  = (const float*)d_in[2];
  const float* B1  = (const float*)d_in[3];
  const float* B2  = (const float*)d_in[4];
  const float* L0  = (const float*)d_in[5];
  const float* L1l = (const float*)d_in[6];
  const float* L1u = (const float*)d_in[7];
  const float* L2  = (const float*)d_in[8];
  const float* D1  = (const float*)d_in[9];
  const float* D2  = (const float*)d_in[10];
  const float* D3  = (const float*)d_in[11];
  const float* D5  = (const float*)d_in[12];
  const float* W0  = (const float*)d_in[13];
  const float* W1  = (const float*)d_in[14];
  const float* W2  = (const float*)d_in[15];
  float* out = (float*)d_out;

  // ---- workspace carve-up (fp32) ----
  float* ws = (float*)d_ws;
  float* d1_inv = ws; ws += N0;
  float* d2v    = ws; ws += N1;
  float* d3v    = ws; ws += N2;
  float* d5p    = ws; ws += N1;
  float* xs0    = ws; ws += (size_t)N0 * 32;        // d1_inv . x0
  float* xs1    = ws; ws += (size_t)N1 * 32;        // d5_pinv . x1
  float* xs2    = ws; ws += (size_t)N2 * 32;        // d3 . x2
  float* X0b    = ws; ws += (size_t)6  * N0 * 32;   // 6 stacked terms, order 0
  float* X1b    = ws; ws += (size_t)11 * N1 * 32;   // 11 terms, order 1
  float* X2b    = ws; ws += (size_t)6  * N2 * 32;   // 6 terms, order 2
  auto X0t = [&](int k) { return X0b + (size_t)k * N0 * 32; };
  auto X1t = [&](int k) { return X1b + (size_t)k * N1 * 32; };
  auto X2t = [&](int k) { return X2b + (size_t)k * N2 * 32; };

  // ---- diagonal extraction ----
  hipLaunchKernelGGL(sccnn_diag_kernel, dim3((N0 + 255) / 256), dim3(256), 0, stream, D1, d1_inv, N0, 1);
  hipLaunchKernelGGL(sccnn_diag_kernel, dim3((N1 + 255) / 256), dim3(256), 0, stream, D2, d2v,    N1, 0);
  hipLaunchKernelGGL(sccnn_diag_kernel, dim3((N2 + 255) / 256), dim3(256), 0, stream, D3, d3v,    N2, 0);
  hipLaunchKernelGGL(sccnn_diag_kernel, dim3((N1 + 255) / 256), dim3(256), 0, stream, D5, d5p,    N1, 2);

  // ---- pre-scaled RHS copies (tiny: <= 768KB each) ----
  hipLaunchKernelGGL(sccnn_rowscale_kernel, dim3((N0 * 32 + 255) / 256), dim3(256), 0, stream, x0, d1_inv, xs0, N0);
  hipLaunchKernelGGL(sccnn_rowscale_kernel, dim3((N1 * 32 + 255) / 256), dim3(256), 0, stream, x1, d5p,    xs1, N1);
  hipLaunchKernelGGL(sccnn_rowscale_kernel, dim3((N2 * 32 + 255) / 256), dim3(256), 0, stream, x2, d3v,    xs2, N2);

  // ---- identity terms: copy raw signals into their stack slots ----
  hipMemcpyAsync(X0t(0), x0, (size_t)N0 * 32 * sizeof(float), hipMemcpyDeviceToDevice, stream);
  hipMemcpyAsync(X1t(3), x1, (size_t)N1 * 32 * sizeof(float), hipMemcpyDeviceToDevice, stream);
  hipMemcpyAsync(X2t(3), x2, (size_t)N2 * 32 * sizeof(float), hipMemcpyDeviceToDevice, stream);

  // ---- boundary projections ----
  // x0p = d1_inv * (B1 @ x1)                 -> X0 slot 3
  run_gemm<0, false>(B1, x1,  X0t(3), d1_inv, N0, N1, stream);
  // x1n = d2 * (B1^T @ (d1_inv * x0))        -> X1 slot 0
  run_gemm<0, true >(B1, xs0, X1t(0), d2v,    N1, N0, stream);
  // x1p = B2 @ (d3 * x2)                     -> X1 slot 8
  run_gemm<0, false>(B2, xs2, X1t(8), nullptr, N1, N2, stream);
  // x2n = B2^T @ (d5_pinv * x1)              -> X2 slot 0
  run_gemm<0, true >(B2, xs1, X2t(0), nullptr, N2, N1, stream);

  // ---- Chebyshev stacks: T1 = L@x ; T2 = 2*(L@T1) - x ----
  run_gemm<0, false>(L0, x0,     X0t(1), nullptr, N0, N0, stream);
  run_gemm<1, false>(L0, X0t(1), X0t(2), x0,      N0, N0, stream);
  run_gemm<0, false>(L0, X0t(3), X0t(4), nullptr, N0, N0, stream);
  run_gemm<1, false>(L0, X0t(4), X0t(5), X0t(3),  N0, N0, stream);

  run_gemm<0, false>(L1l, X1t(0), X1t(1), nullptr, N1, N1, stream);
  run_gemm<1, false>(L1l, X1t(1), X1t(2), X1t(0),  N1, N1, stream);
  run_gemm<0, false>(L1l, x1,     X1t(4), nullptr, N1, N1, stream);
  run_gemm<1, false>(L1l, X1t(4), X1t(5), x1,      N1, N1, stream);

  run_gemm<0, false>(L1u, x1,     X1t(6),  nullptr, N1, N1, stream);
  run_gemm<1, false>(L1u, X1t(6), X1t(7),  x1,      N1, N1, stream);
  run_gemm<0, false>(L1u, X1t(8), X1t(9),  nullptr, N1, N1, stream);
  run_gemm<1, false>(L1u, X1t(9), X1t(10), X1t(8),  N1, N1, stream);

  run_gemm<0, false>(L2, X2t(0), X2t(1), nullptr, N2, N2, stream);
  run_gemm<1, false>(L2, X2t(1), X2t(2), X2t(0),  N2, N2, stream);
  run_gemm<0, false>(L2, x2,     X2t(4), nullptr, N2, N2, stream);
  run_gemm<1, false>(L2, X2t(4), X2t(5), x2,      N2, N2, stream);

  // ---- fused per-node einsum + ReLU ----
  hipLaunchKernelGGL(sccnn_fused_out_kernel, dim3((N0 + 7) / 8), dim3(NTHREADS), 0, stream,
                     X0b, W0, out, N0, 6);
  hipLaunchKernelGGL(sccnn_fused_out_kernel, dim3((N1 + 7) / 8), dim3(NTHREADS), 0, stream,
                     X1b, W1, out + (size_t)N0 * 32, N1, 11);
  hipLaunchKernelGGL(sccnn_fused_out_kernel, dim3((N2 + 7) / 8), dim3(NTHREADS), 0, stream,
                     X2b, W2, out + (size_t)(N0 + N1) * 32, N2, 6);
}